// Decoder_37374805410006
// MI455X (gfx1250) — compile-verified
//
#include <hip/hip_runtime.h>

// Conv-TasNet decoder tail: windowed overlap-and-add, hop=8, frame=16, trim 8 each side.
// out[b,c,m] = x[b,c,l,t1]*w[b,l,t1] + x[b,c,l+8,t1-1]*w[b,l+8,t1-1]
//   with t1 = m/8 + 1, l = m%8.  Pure streaming: 262 MB traffic, ~11.3 us @ 23.3 TB/s.

typedef float v4f __attribute__((ext_vector_type(4)));

constexpr int B  = 8;
constexpr int C  = 2;
constexpr int D  = 16;
constexpr int T  = 128000;
constexpr int NT = T - 1;        // 127999 output groups of 8 per (b,c)
constexpr int M  = 8 * NT;       // 1023992 samples per (b,c)

__global__ __launch_bounds__(256) void oa_decoder_kernel(
    const float* __restrict__ x,   // [B, C, D, T]
    const float* __restrict__ w,   // [B, D, T]
    float* __restrict__ out)       // [B, C, M]
{
    const int tid = blockIdx.x * blockDim.x + threadIdx.x;
    if (tid >= B * NT) return;

    const int b  = tid / NT;
    const int t1 = tid - b * NT + 1;   // 1 .. T-1, lane-consecutive -> coalesced
    const int t0 = t1 - 1;

    const float* __restrict__ wb  = w + (size_t)b * D * T;
    const float* __restrict__ xc0 = x + ((size_t)b * C + 0) * (size_t)D * T;
    const float* __restrict__ xc1 = x + ((size_t)b * C + 1) * (size_t)D * T;

    float o0[8], o1[8];
#pragma unroll
    for (int l = 0; l < 8; ++l) {
        // Every element below is touched exactly once across the whole grid:
        // rows l (0..7) are consumed at t=t1, rows l+8 at t=t0. Use NT hints
        // so this one-shot stream does not pollute L2.
        const float w1 = __builtin_nontemporal_load(wb  + (size_t)l       * T + t1);
        const float w0 = __builtin_nontemporal_load(wb  + (size_t)(l + 8) * T + t0);
        const float a1 = __builtin_nontemporal_load(xc0 + (size_t)l       * T + t1);
        const float a0 = __builtin_nontemporal_load(xc0 + (size_t)(l + 8) * T + t0);
        const float b1 = __builtin_nontemporal_load(xc1 + (size_t)l       * T + t1);
        const float b0 = __builtin_nontemporal_load(xc1 + (size_t)(l + 8) * T + t0);
        o0[l] = fmaf(a1, w1, a0 * w0);
        o1[l] = fmaf(b1, w1, b0 * w0);
    }

    // 8 contiguous outputs per channel starting at m = 8*t0; 16B-aligned since
    // M % 8 == 0 and m % 8 == 0  ->  two global_store_b128 per channel.
    const size_t m  = (size_t)8 * (size_t)t0;
    float* p0 = out + ((size_t)b * C + 0) * (size_t)M + m;
    float* p1 = out + ((size_t)b * C + 1) * (size_t)M + m;

    const v4f q00 = {o0[0], o0[1], o0[2], o0[3]};
    const v4f q01 = {o0[4], o0[5], o0[6], o0[7]};
    const v4f q10 = {o1[0], o1[1], o1[2], o1[3]};
    const v4f q11 = {o1[4], o1[5], o1[6], o1[7]};

    __builtin_nontemporal_store(q00, (v4f*)p0);
    __builtin_nontemporal_store(q01, (v4f*)(p0 + 4));
    __builtin_nontemporal_store(q10, (v4f*)p1);
    __builtin_nontemporal_store(q11, (v4f*)(p1 + 4));
}

extern "C" void kernel_launch(void* const* d_in, const int* in_sizes, int n_in,
                              void* d_out, int out_size, void* d_ws, size_t ws_size,
                              hipStream_t stream)
{
    (void)in_sizes; (void)n_in; (void)out_size; (void)d_ws; (void)ws_size;
    const float* x   = (const float*)d_in[0];   // [B, C, D, T] fp32
    const float* w   = (const float*)d_in[1];   // [B, D, T]    fp32
    float*       out = (float*)d_out;           // [B, C, M]    fp32
    // d_in[2], d_in[3] are pad_left/pad_right == 8, baked into the index math.

    const int total  = B * NT;                  // 1,023,992 threads
    const int block  = 256;                     // 8 wave32 waves
    const int grid   = (total + block - 1) / block;  // 4000 blocks
    hipLaunchKernelGGL(oa_decoder_kernel, dim3(grid), dim3(block), 0, stream,
                       x, w, out);
}